// MultiChannel2DCircularConv_84052509982884
// MI455X (gfx1250) — compile-verified
//
#include <hip/hip_runtime.h>
#include <math.h>

#define BATCH 32
#define CH    96
#define NDIM  64
#define KW    3
#define PIX   (NDIM*NDIM)      // 4096
#define WSTR  (CH*KW*KW)       // 864 floats per output channel of the kernel tensor
#define WPACK (9*CH*CH)        // repacked weights: [tap][ci][co] = 82944 floats

typedef __attribute__((ext_vector_type(2))) float v2f;
typedef __attribute__((ext_vector_type(8))) float v8f;
typedef __attribute__((ext_vector_type(4))) unsigned int v4u;
typedef __attribute__((ext_vector_type(8))) int v8i;
typedef __attribute__((ext_vector_type(4))) int v4i;

// ---------------------------------------------------------------------------
// Repack kernel weights w[co][ci][ky][kx] -> wp[tap][ci][co] so that
// (a) WMMA A-fragment reads are contiguous in co, and
// (b) a K-chunk slice is a rectangular 3D tile the TDM can DMA into LDS.
// ---------------------------------------------------------------------------
__global__ void __launch_bounds__(256)
repack_w_kernel(const float* __restrict__ w, float* __restrict__ wp)
{
    const int idx = blockIdx.x * 256 + threadIdx.x;
    if (idx >= WPACK) return;
    const int t   = idx / (CH*CH);
    const int rem = idx - t * (CH*CH);
    const int ci  = rem / CH;
    const int co  = rem - ci * CH;
    wp[idx] = w[(size_t)co * WSTR + (size_t)ci * (KW*KW) + t];
}

// ---------------------------------------------------------------------------
// Conv as 9 tap-shifted GEMMs with fp32 WMMA (V_WMMA_F32_16X16X4_F32).
// grid = (16 row-blocks, 32 batches), block = 256 threads (8 waves).
// Per K-chunk of 16 input channels:
//   - wave 0 issues a TDM 3D-tile load of the 55KB weight slice into LDS
//   - all waves stage the wrap-padded input tile (16ch x 6rows x 64cols)
//   - s_wait_tensorcnt + barrier, then 9 taps x 4 K-steps x 12 tiles of WMMA.
// ---------------------------------------------------------------------------
__global__ void __launch_bounds__(256)
conv_wmma_kernel(const float* __restrict__ x,
                 const float* __restrict__ wp,
                 const float* __restrict__ bias,
                 float* __restrict__ out)
{
    __shared__ float lds_in[16 * 6 * 64];   // [ch][row][col], 24576 B
    __shared__ float w_tile[9 * 16 * 96];   // [tap][ci][co], 55296 B

    const int b    = blockIdx.y;
    const int r0   = blockIdx.x * 4;        // first output row of this block
    const int tid  = threadIdx.x;
    const int lane = tid & 31;
    const int wv   = tid >> 5;              // wave id 0..7

    const int n    = lane & 15;             // N index inside 16x16 tile
    const int hi   = lane >> 4;             // 0 / 1 half of wave
    const int koff = hi * 2;                // K offset held by this lane (fp32 WMMA layout)

    v8f acc[2][6];
    const v8f vzero = {0.f,0.f,0.f,0.f,0.f,0.f,0.f,0.f};
#pragma unroll
    for (int pi = 0; pi < 2; ++pi)
#pragma unroll
        for (int ct = 0; ct < 6; ++ct) acc[pi][ct] = vzero;

    for (int cc = 0; cc < 6; ++cc) {        // K-chunks of 16 input channels
        const int ci0 = cc * 16;

        // ---- wave 0: TDM async DMA of weight slice wp[0..8][ci0..ci0+15][0..95] ----
        if (wv == 0) {
            const unsigned long long ga =
                (unsigned long long)(uintptr_t)(wp + (size_t)ci0 * CH);
            const unsigned int lds_base =
                (unsigned int)(uintptr_t)(&w_tile[0]);      // LDS aperture low bits
            v4u g0;
            g0.x = 1u;                                       // count = 1 descriptor
            g0.y = lds_base;                                 // lds_addr
            g0.z = (unsigned int)(ga & 0xffffffffu);         // global_addr[31:0]
            g0.w = (unsigned int)((ga >> 32) & 0x1ffffffu)   // global_addr[56:32]
                   | (2u << 30);                             // type = 2 (image)
            v8i g1;
            g1[0] = (int)(2u << 16);                         // data_size = 4B
            g1[1] = (int)(96u << 16);                        // tensor_dim0 = 96 (lo16)
            g1[2] = (int)(96u << 16);                        // dim0 hi=0 | tensor_dim1=96 lo16
            g1[3] = (int)(96u << 16);                        // dim1 hi=0 | tile_dim0 = 96
            g1[4] = (int)(16u | (9u << 16));                 // tile_dim1=16, tile_dim2=9
            g1[5] = 96;                                      // tensor_dim0_stride lo32
            g1[6] = (int)(9216u << 16);                      // s0 hi=0 | tensor_dim1_stride lo16
            g1[7] = 0;                                       // tensor_dim1_stride hi
            v4i g2;
            g2[0] = 9;                                       // tensor_dim2 = 9 taps
            g2[1] = 0;                                       // tensor_dim3
            g2[2] = 9216;                                    // tensor_dim2_stride lo32
            g2[3] = 0;                                       // stride hi | tile_dim3 = 0
            v4i g3 = {0, 0, 0, 0};
            v8i g4 = {0, 0, 0, 0, 0, 0, 0, 0};               // extra group (6-arg form)
            __builtin_amdgcn_tensor_load_to_lds(g0, g1, g2, g3, g4, 0);
        }

        // ---- all waves: stage (x + bias) with circular row halo (float4) ----
#pragma unroll
        for (int j = 0; j < 6; ++j) {       // 1536 float4 elements / 256 threads
            const int e   = tid + 256 * j;
            const int ch  = e / 96;
            const int rem = e - ch * 96;
            const int r   = rem >> 4;
            const int c4  = rem & 15;
            const int grow = (r0 - 1 + r + NDIM) & (NDIM - 1);
            const size_t goff = (size_t)grow * NDIM + (size_t)c4 * 4;
            const float4 xv = *(const float4*)(x    + (size_t)(b*CH + ci0 + ch) * PIX + goff);
            const float4 bv = *(const float4*)(bias + (size_t)(ci0 + ch)        * PIX + goff);
            float4 s; s.x = xv.x+bv.x; s.y = xv.y+bv.y; s.z = xv.z+bv.z; s.w = xv.w+bv.w;
            *(float4*)(&lds_in[ch*384 + r*64 + c4*4]) = s;
        }

        if (wv == 0) __builtin_amdgcn_s_wait_tensorcnt(0);
        __syncthreads();

#pragma unroll
        for (int t = 0; t < 9; ++t) {       // 3x3 taps (fully unrolled: const dy/dx)
            const int dy = t / 3;
            const int dx = t - 3 * dy;
#pragma unroll
            for (int kb = 0; kb < 4; ++kb) {    // K sub-steps of 4 inside chunk
                const int ci_l = kb * 4 + koff;     // per-lane channel in chunk
                // A fragments from LDS: w_tile[t][ci_l(+1)][ct*16+n]
                v2f a[6];
#pragma unroll
                for (int ct = 0; ct < 6; ++ct) {
                    a[ct].x = w_tile[(t*16 + ci_l    ) * 96 + ct*16 + n];
                    a[ct].y = w_tile[(t*16 + ci_l + 1) * 96 + ct*16 + n];
                }
#pragma unroll
                for (int pi = 0; pi < 2; ++pi) {
                    const int pt   = wv + 8 * pi;               // pixel tile 0..15
                    const int ldr  = (pt >> 2) + dy;            // LDS row 0..5
                    const int col  = (((pt & 3)*16 + n) + dx - 1 + NDIM) & (NDIM - 1);
                    v2f bf;
                    bf.x = lds_in[ ci_l      * 384 + ldr*64 + col];
                    bf.y = lds_in[(ci_l + 1) * 384 + ldr*64 + col];
#pragma unroll
                    for (int ct = 0; ct < 6; ++ct) {
                        acc[pi][ct] = __builtin_amdgcn_wmma_f32_16x16x4_f32(
                            false, a[ct], false, bf, (short)0, acc[pi][ct], false, false);
                    }
                }
            }
        }
        __syncthreads();
    }

    // ---- store: C/D layout (vgpr v -> M = v + 8*hi, lane&15 -> N) ----
#pragma unroll
    for (int pi = 0; pi < 2; ++pi) {
        const int pt   = wv + 8 * pi;
        const int orow = r0 + (pt >> 2);
        const int ocol = (pt & 3) * 16 + n;
#pragma unroll
        for (int ct = 0; ct < 6; ++ct) {
#pragma unroll
            for (int v = 0; v < 8; ++v) {
                const int co = ct*16 + v + 8*hi;
                out[(size_t)(b*CH + co) * PIX + (size_t)orow * NDIM + ocol] = acc[pi][ct][v];
            }
        }
    }
}

// ---------------------------------------------------------------------------
// logdet: one workgroup per frequency bin (u,v). Build K_hat (96x96 complex)
// in LDS from the 9 kernel taps, LU with partial pivoting, sum log|pivot|.
// ---------------------------------------------------------------------------
__global__ void zero_scalar_kernel(float* p)
{
    if (threadIdx.x == 0 && blockIdx.x == 0) *p = 0.f;
}

__global__ void __launch_bounds__(128)
logdet_kernel(const float* __restrict__ w, float* __restrict__ out_scalar)
{
    __shared__ float2 M[CH * CH];   // 96*96 complex fp32 = 73728 B (LDS is 320KB/WGP)
    __shared__ int    s_piv;

    const int bin = blockIdx.x;
    const int u   = bin >> 6;
    const int v   = bin & 63;
    const int tid = threadIdx.x;

    // twiddles for the 3x3 taps: exp(-2*pi*i*(u*a + v*b)/64)
    float twr[9], twi[9];
#pragma unroll
    for (int a = 0; a < 3; ++a)
#pragma unroll
        for (int bb = 0; bb < 3; ++bb) {
            const float ang = -2.f * 3.14159265358979323846f
                              * (float)(u*a + v*bb) / (float)NDIM;
            twr[a*3+bb] = cosf(ang);
            twi[a*3+bb] = sinf(ang);
        }

    // build K_hat
    for (int idx = tid; idx < CH*CH; idx += 128) {
        const int o = idx / CH;
        const int i = idx - o * CH;
        const float* kp = w + (size_t)o * WSTR + (size_t)i * (KW*KW);
        float re = 0.f, im = 0.f;
#pragma unroll
        for (int t = 0; t < 9; ++t) {
            const float kv = kp[t];
            re += kv * twr[t];
            im += kv * twi[t];
        }
        M[idx] = make_float2(re, im);
    }
    __syncthreads();

    float logacc = 0.f;
    for (int k = 0; k < CH; ++k) {
        if (tid == 0) {
            int   piv  = k;
            float best = M[k*CH+k].x*M[k*CH+k].x + M[k*CH+k].y*M[k*CH+k].y;
            for (int j = k + 1; j < CH; ++j) {
                const float2 e = M[j*CH+k];
                const float m2 = e.x*e.x + e.y*e.y;
                if (m2 > best) { best = m2; piv = j; }
            }
            s_piv = piv;
        }
        __syncthreads();
        const int piv = s_piv;
        if (piv != k) {
            for (int c = k + tid; c < CH; c += 128) {
                const float2 tmp = M[k*CH+c];
                M[k*CH+c]   = M[piv*CH+c];
                M[piv*CH+c] = tmp;
            }
        }
        __syncthreads();
        const float2 p   = M[k*CH+k];
        const float  den = p.x*p.x + p.y*p.y;
        if (tid == 0) logacc += 0.5f * logf(den);
        const float inv = 1.f / den;
        for (int j = k + 1 + tid; j < CH; j += 128) {
            const float2 a = M[j*CH+k];
            const float fr = (a.x*p.x + a.y*p.y) * inv;   // a / p = a*conj(p)/|p|^2
            const float fi = (a.y*p.x - a.x*p.y) * inv;
            for (int c = k + 1; c < CH; ++c) {
                const float2 kc = M[k*CH+c];
                float2 jc = M[j*CH+c];
                jc.x -= fr*kc.x - fi*kc.y;
                jc.y -= fr*kc.y + fi*kc.x;
                M[j*CH+c] = jc;
            }
            M[j*CH+k] = make_float2(0.f, 0.f);
        }
        __syncthreads();
    }
    if (tid == 0) atomicAdd(out_scalar, logacc);
}

// ---------------------------------------------------------------------------
extern "C" void kernel_launch(void* const* d_in, const int* in_sizes, int n_in,
                              void* d_out, int out_size, void* d_ws, size_t ws_size,
                              hipStream_t stream)
{
    (void)in_sizes; (void)n_in; (void)ws_size; (void)out_size;
    const float* x    = (const float*)d_in[0];
    const float* w    = (const float*)d_in[1];
    const float* bias = (const float*)d_in[2];
    float* out        = (float*)d_out;
    float* logdet_out = out + (size_t)BATCH * CH * PIX;   // last element
    float* w_packed   = (float*)d_ws;                     // 331776 B scratch

    repack_w_kernel<<<(WPACK + 255) / 256, 256, 0, stream>>>(w, w_packed);
    conv_wmma_kernel<<<dim3(16, 32), 256, 0, stream>>>(x, w_packed, bias, out);
    zero_scalar_kernel<<<1, 32, 0, stream>>>(logdet_out);
    logdet_kernel<<<PIX, 128, 0, stream>>>(w, logdet_out);
}